// ClusterFlipModule_67851893342541
// MI455X (gfx1250) — compile-verified
//
#include <hip/hip_runtime.h>
#include <hip/hip_bf16.h>

#define NBINS 4096

typedef __attribute__((ext_vector_type(2))) float v2f;
typedef __attribute__((ext_vector_type(4))) float f4;
typedef __attribute__((ext_vector_type(8))) float v8f;

// ---------------------------------------------------------------------------
// K0: zero histogram + threshold scratch (ws is not re-poisoned between calls)
// ---------------------------------------------------------------------------
__global__ void k0_zero(int* __restrict__ p, int count) {
    int i = blockIdx.x * blockDim.x + threadIdx.x;
    if (i < count) p[i] = 0;
}

// ---------------------------------------------------------------------------
// K1: labels + MLP importance. Layer-1 (2->8) done with V_WMMA_F32_16X16X4_F32:
//   A (16x4 f32): 16 points, K = {fx, fy, 0, 0}
//     lanes 0-15 hold {K0,K1} = {fx,fy}; lanes 16-31 hold {K2,K3} = {0,0}
//   B (4x16 f32): w1 padded (cols 8..15 zero); lanes 16-31 = K2,K3 rows = 0
//   D (16x16 f32): row M = point, col N = hidden unit (N<8 valid)
// Layer-2 (8->1): per-lane partial relu(d+b1[N])*w2[N], 16-lane shuffle reduce.
// One wave handles 16 points.
// ---------------------------------------------------------------------------
__global__ void k1_score(const float* __restrict__ features,
                         const float* __restrict__ centers,
                         const float* __restrict__ w1, const float* __restrict__ b1,
                         const float* __restrict__ w2, const float* __restrict__ b2,
                         float* __restrict__ imp, int* __restrict__ label, int n)
{
    int gtid = blockIdx.x * blockDim.x + threadIdx.x;
    int wave = gtid >> 5;
    int lane = gtid & 31;
    int p0 = wave * 16;
    if (p0 >= n) return;                 // wave-uniform exit

    int half = lane >> 4;                // 0: K=0,1   1: K=2,3
    int mrow = lane & 15;                // A: row M, B/D: col N

    v2f a; a[0] = 0.f; a[1] = 0.f;
    float fx = 0.f, fy = 0.f;
    if (half == 0) {
        int p = p0 + mrow;
        fx = features[2 * p + 0];
        fy = features[2 * p + 1];
        a[0] = fx; a[1] = fy;
    }
    v2f bm; bm[0] = 0.f; bm[1] = 0.f;
    if (half == 0 && mrow < 8) {         // w1 row-major (2,8): w1[k*8 + j]
        bm[0] = w1[0 * 8 + mrow];
        bm[1] = w1[1 * 8 + mrow];
    }

    v8f c = {};
    v8f d = __builtin_amdgcn_wmma_f32_16x16x4_f32(
        /*neg_a=*/false, a, /*neg_b=*/false, bm,
        /*c_mod=*/(short)0, c, /*reuse_a=*/false, /*reuse_b=*/false);

    // This lane owns column N = mrow for rows M = r + 8*half (r = VGPR index).
    int   N  = mrow;
    float wn = (N < 8) ? w2[N] : 0.f;
    float bn = (N < 8) ? b1[N] : 0.f;

    float s[8];
#pragma unroll
    for (int r = 0; r < 8; ++r) {
        float h = d[r] + bn;
        h = h > 0.f ? h : 0.f;           // ReLU
        s[r] = (N < 8) ? h * wn : 0.f;   // layer-2 partial
    }
    // reduce across the 16 lanes of each half (xor offsets stay within halves)
#pragma unroll
    for (int r = 0; r < 8; ++r) {
#pragma unroll
        for (int off = 1; off < 16; off <<= 1)
            s[r] += __shfl_xor(s[r], off, 32);
    }
    if (mrow == 0) {                     // lanes 0 and 16 write 8 points each
        float bias2 = b2[0];
#pragma unroll
        for (int r = 0; r < 8; ++r) {
            int p = p0 + r + 8 * half;
            float z = s[r] + bias2;
            imp[p] = 1.0f / (1.0f + __expf(-z));
        }
    }
    // Labels: lanes 0-15 handle point p0+mrow (argmin of squared distance)
    if (half == 0) {
        int best = 0; float bd = 3.0e38f;
#pragma unroll
        for (int cc = 0; cc < 3; ++cc) {
            float dx = fx - centers[2 * cc + 0];
            float dy = fy - centers[2 * cc + 1];
            float dd = dx * dx + dy * dy;
            if (dd < bd) { bd = dd; best = cc; }
        }
        label[p0 + mrow] = best;
    }
}

// ---------------------------------------------------------------------------
// K2: per-cluster histogram of importance (sigmoid -> bounded in (0,1))
// ---------------------------------------------------------------------------
__global__ void k2_hist(const float* __restrict__ imp, const int* __restrict__ label,
                        int* __restrict__ hist, int n)
{
    int i = blockIdx.x * blockDim.x + threadIdx.x;
    if (i < n) {
        int bin = (int)(imp[i] * (float)NBINS);
        bin = bin < 0 ? 0 : (bin > NBINS - 1 ? NBINS - 1 : bin);
        atomicAdd(&hist[label[i] * NBINS + bin], 1);
    }
}

// ---------------------------------------------------------------------------
// K3: per-cluster rank-threshold bins. 3 waves, each warp-scans its cluster's
// 4096-bin CDF (32 bins/step, shuffle prefix sum) from the top for bin_hi
// (top num_high) and from the bottom for bin_lo (bottom num_low).
// ---------------------------------------------------------------------------
__global__ void k3_thresh(const int* __restrict__ hist, int* __restrict__ thr,
                          const int* __restrict__ epoch,
                          const int* __restrict__ max_epochs, int n)
{
    int lane = threadIdx.x & 31;
    int wave = threadIdx.x >> 5;
    if (wave >= 3) return;

    float progress = (float)epoch[0] / (float)max_epochs[0];
    int num_high = (int)(0.5f * (1.0f - progress) * (float)n);
    int num_low  = (int)(0.5f * progress * (float)n);
    if (num_high < 1) num_high = 1;
    if (num_low  < 1) num_low  = 1;

    const int* h = hist + wave * NBINS;

    // --- scan from top: highest bin where cum-from-top >= num_high ---
    int running = 0;
    int bin_hi = 0;
    for (int chunk = 0; chunk < NBINS / 32; ++chunk) {
        int bin = NBINS - 1 - (chunk * 32 + lane);
        int s = h[bin];
#pragma unroll
        for (int off = 1; off < 32; off <<= 1) {
            int t = __shfl_up(s, off, 32);
            if (lane >= off) s += t;
        }
        unsigned long long m = __ballot(running + s >= num_high);
        if (m) {
            int first = __ffsll((long long)m) - 1;
            bin_hi = NBINS - 1 - (chunk * 32 + first);
            break;
        }
        running += __shfl(s, 31, 32);
    }

    // --- scan from bottom: lowest bin where cum-from-bottom >= num_low ---
    running = 0;
    int bin_lo = NBINS - 1;
    for (int chunk = 0; chunk < NBINS / 32; ++chunk) {
        int bin = chunk * 32 + lane;
        int s = h[bin];
#pragma unroll
        for (int off = 1; off < 32; off <<= 1) {
            int t = __shfl_up(s, off, 32);
            if (lane >= off) s += t;
        }
        unsigned long long m = __ballot(running + s >= num_low);
        if (m) {
            int first = __ffsll((long long)m) - 1;
            bin_lo = chunk * 32 + first;
            break;
        }
        running += __shfl(s, 31, 32);
    }

    if (lane == 0) {
        thr[wave * 2 + 0] = bin_hi;
        thr[wave * 2 + 1] = bin_lo;
    }
}

// ---------------------------------------------------------------------------
// K4: bandwidth kernel. One row (512 f32) per block; float4 (b128) traffic in
// both directions; reversed reads are still wave-contiguous -> coalesced.
// Non-temporal hints: 512 MB stream has no reuse, keep it out of caches.
// flip = in_top_k XOR in_bottom_k (bin-granular thresholds).
// ---------------------------------------------------------------------------
__global__ void k4_flip(const float* __restrict__ blocks,
                        const float* __restrict__ imp,
                        const int* __restrict__ label,
                        const int* __restrict__ thr,
                        float* __restrict__ out, int blen)
{
    int row = blockIdx.x;
    int c   = label[row];
    int bin = (int)(imp[row] * (float)NBINS);
    bin = bin < 0 ? 0 : (bin > NBINS - 1 ? NBINS - 1 : bin);
    bool in_high = bin >= thr[c * 2 + 0];
    bool in_low  = bin <= thr[c * 2 + 1];
    bool flip = in_high ^ in_low;

    int vec = blen >> 2;                               // float4s per row (128)
    const f4* src = (const f4*)(blocks + (size_t)row * blen);
    f4*       dst = (f4*)(out + (size_t)row * blen);

    for (int t = threadIdx.x; t < vec; t += blockDim.x) {
        if (flip) {
            f4 x = __builtin_nontemporal_load(&src[vec - 1 - t]);
            f4 r; r[0] = x[3]; r[1] = x[2]; r[2] = x[1]; r[3] = x[0];
            __builtin_nontemporal_store(r, &dst[t]);
        } else {
            f4 x = __builtin_nontemporal_load(&src[t]);
            __builtin_nontemporal_store(x, &dst[t]);
        }
    }
}

// ---------------------------------------------------------------------------
extern "C" void kernel_launch(void* const* d_in, const int* in_sizes, int n_in,
                              void* d_out, int out_size, void* d_ws, size_t ws_size,
                              hipStream_t stream) {
    const float* features   = (const float*)d_in[0];   // (n, 2)
    const float* blocks     = (const float*)d_in[1];   // (n, 512)
    const float* centers    = (const float*)d_in[2];   // (3, 2)
    const float* w1         = (const float*)d_in[3];   // (2, 8)
    const float* b1         = (const float*)d_in[4];   // (8,)
    const float* w2         = (const float*)d_in[5];   // (8, 1)
    const float* b2         = (const float*)d_in[6];   // (1,)
    const int*   epoch      = (const int*)d_in[7];
    const int*   max_epochs = (const int*)d_in[8];

    int n    = in_sizes[0] / 2;            // 131072
    int blen = in_sizes[1] / n;            // 512

    // workspace layout
    float* imp   = (float*)d_ws;                                   // n floats
    int*   label = (int*)((char*)d_ws + (size_t)n * 4);            // n ints
    int*   hist  = (int*)((char*)d_ws + (size_t)n * 8);            // 3*NBINS
    int*   thr   = hist + 3 * NBINS;                               // 8 ints

    k0_zero<<<(3 * NBINS + 8 + 255) / 256, 256, 0, stream>>>(hist, 3 * NBINS + 8);

    int waves = n / 16;                    // 16 points per wave
    int total = waves * 32;
    k1_score<<<(total + 255) / 256, 256, 0, stream>>>(
        features, centers, w1, b1, w2, b2, imp, label, n);

    k2_hist<<<(n + 255) / 256, 256, 0, stream>>>(imp, label, hist, n);

    k3_thresh<<<1, 128, 0, stream>>>(hist, thr, epoch, max_epochs, n);

    k4_flip<<<n, 128, 0, stream>>>(blocks, imp, label, thr, (float*)d_out, blen);
}